// CTC_72000831750194
// MI455X (gfx1250) — compile-verified
//
#include <hip/hip_runtime.h>
#include <stdint.h>

#define BB 32
#define TT 1000
#define VV 1024
#define LL 128
#define SS 257          // 2L+1 extended lattice width
#define SP 264          // padded row (multiple of 8 floats = 32B per lane)
#define NEGF (-1e30f)

typedef float    v2f  __attribute__((ext_vector_type(2)));
typedef float    v8f  __attribute__((ext_vector_type(8)));
typedef uint32_t u32x4 __attribute__((ext_vector_type(4)));
typedef uint32_t u32x8 __attribute__((ext_vector_type(8)));

// ---------------------------------------------------------------------------
// Phase 1+2 fused: per-(b,t) logsumexp + gather of extended-label log-probs.
// One wave32 per row; exp-sum reduction runs on the matrix pipe via
// V_WMMA_F32_16X16X4_F32 with an all-ones B matrix.
// ---------------------------------------------------------------------------
__global__ __launch_bounds__(256) void ctc_lse_gather(
    const float* __restrict__ hs, const int* __restrict__ ys,
    float* __restrict__ lp, float* __restrict__ skipm)
{
  const int lane = threadIdx.x & 31;
  const int wave = threadIdx.x >> 5;
  const int row  = blockIdx.x * 8 + wave;       // row = b*TT + t
  const int b    = row / TT;
  const int t    = row - b * TT;
  const float* __restrict__ base = hs + (size_t)row * VV;

  // 32 elements per lane, kept in registers (16 x float2, coalesced loads)
  v2f v[16];
  float m = NEGF;
#pragma unroll
  for (int i = 0; i < 16; ++i) {
    v[i] = *(const v2f*)(base + 64 * i + 2 * lane);
    m = fmaxf(m, fmaxf(v[i][0], v[i][1]));
  }
#pragma unroll
  for (int off = 16; off >= 1; off >>= 1)
    m = fmaxf(m, __shfl_xor(m, off, 32));

  // sum(exp(x-m)) via fp32 WMMA: A[m,k] holds exps, B = ones(4x16)
  // D[m,n] = row-sum replicated over n; accumulate over 16 chunks.
  v8f c0 = {}; v8f c1 = {};
  v2f ones; ones[0] = 1.0f; ones[1] = 1.0f;
#pragma unroll
  for (int i = 0; i < 16; i += 2) {
    v2f e0; e0[0] = __expf(v[i][0] - m);     e0[1] = __expf(v[i][1] - m);
    v2f e1; e1[0] = __expf(v[i + 1][0] - m); e1[1] = __expf(v[i + 1][1] - m);
    c0 = __builtin_amdgcn_wmma_f32_16x16x4_f32(false, e0, false, ones,
                                               (short)0, c0, false, false);
    c1 = __builtin_amdgcn_wmma_f32_16x16x4_f32(false, e1, false, ones,
                                               (short)0, c1, false, false);
  }
  float s = 0.0f;
#pragma unroll
  for (int k = 0; k < 8; ++k) s += c0[k] + c1[k];   // rows 0-7 (lanes<16) / 8-15
  s += __shfl_xor(s, 16, 32);                       // total over all 1024
  const float lse = m + __logf(s);

  // gather lp_ext[b,t,s] = hs[b,t,ext[s]] - lse (row is hot in cache)
  const int* yrow = ys + b * LL;
  float* lrow = lp + (size_t)row * SP;
#pragma unroll
  for (int it = 0; it < 9; ++it) {
    int sidx = 32 * it + lane;
    if (sidx < SP) {
      float out = NEGF;
      if (sidx < SS) {
        int lab = (sidx & 1) ? yrow[(sidx - 1) >> 1] : 0;   // blank at even s
        out = base[lab] - lse;
      }
      lrow[sidx] = out;
    }
  }

  // additive skip mask (0 allowed / NEG blocked), written once per batch
  if (t == 0) {
    float* srow = skipm + b * SP;
#pragma unroll
    for (int it = 0; it < 9; ++it) {
      int sidx = 32 * it + lane;
      if (sidx < SP) {
        float msk = NEGF;
        if (sidx < SS && sidx >= 3 && (sidx & 1)) {
          int lab  = yrow[(sidx - 1) >> 1];
          int lab2 = yrow[(sidx - 3) >> 1];
          if (lab != 0 && lab != lab2) msk = 0.0f;
        }
        srow[sidx] = msk;
      }
    }
  }
}

// ---------------------------------------------------------------------------
// Phase 3: alpha DP. One wave32 per batch element; 257 states blocked
// 8-per-lane (+state 256 on lane 31). Cross-lane neighbors via shfl_up only.
// Per-step lp row prefetched double-buffered into LDS by the TDM
// (tensor_load_to_lds), waited with s_wait_tensorcnt.
// ---------------------------------------------------------------------------
__global__ __launch_bounds__(32) void ctc_alpha_dp(
    const float* __restrict__ lp, const float* __restrict__ skipm,
    const int* __restrict__ hlen, const int* __restrict__ ylen,
    float* __restrict__ lossb)
{
  __shared__ float buf[2][SP];
  __shared__ float afin[SS];
  const int b  = blockIdx.x;
  const int l  = threadIdx.x;          // 0..31
  const int hl = hlen[b];

  float a[9], sk[9];
#pragma unroll
  for (int j = 0; j < 9; ++j) { a[j] = NEGF; sk[j] = NEGF; }
#pragma unroll
  for (int j = 0; j < 8; ++j) sk[j] = skipm[b * SP + 8 * l + j];
  // sk[8]: state 256 (blank) never has a skip transition -> stays NEG

  const float* lrow0 = lp + (size_t)(b * TT) * SP;
  if (l == 0) { a[0] = lrow0[0]; a[1] = lrow0[1]; }   // alpha at t=0

  const uint64_t lp_base = (uint64_t)(uintptr_t)lrow0;

  // TDM D# group1 (constant): data_size=4B, 1-row tile of SP elements
  u32x8 g1;
  g1[0] = 0x00020000u;                       // wg_mask=0, data_size=2 (4B)
  g1[1] = ((uint32_t)SP & 0xffffu) << 16;    // tensor_dim0[15:0]
  g1[2] = ((uint32_t)SP >> 16) | (1u << 16); // tensor_dim0[31:16] | tensor_dim1=1
  g1[3] = ((uint32_t)SP) << 16;              // tile_dim0 = SP
  g1[4] = 1u;                                // tile_dim1=1, tile_dim2=0
  g1[5] = (uint32_t)SP;                      // tensor_dim0_stride (elements)
  g1[6] = 0u;
  g1[7] = 0u;

  // prologue: async-load lp row t=1 into buf[1]
  {
    uint32_t lds_off = (uint32_t)(uintptr_t)&buf[1][0];
    uint64_t ga = lp_base + (uint64_t)SP * 4u;
    u32x4 g0;
    g0[0] = 1u;                              // count=1, user descriptor
    g0[1] = lds_off;                         // lds_addr (bytes)
    g0[2] = (uint32_t)ga;                    // global_addr[31:0]
    g0[3] = (uint32_t)((ga >> 32) & 0x01ffffffu) | (2u << 30); // addr hi | type=2
    asm volatile("tensor_load_to_lds %0, %1" :: "s"(g0), "s"(g1) : "memory");
  }

  for (int t = 1; t < TT; ++t) {
    const int cur = t & 1;
    if (t + 1 < TT) {
      uint32_t lds_off = (uint32_t)(uintptr_t)&buf[(t + 1) & 1][0];
      uint64_t ga = lp_base + (uint64_t)SP * 4u * (uint64_t)(t + 1);
      u32x4 g0;
      g0[0] = 1u;
      g0[1] = lds_off;
      g0[2] = (uint32_t)ga;
      g0[3] = (uint32_t)((ga >> 32) & 0x01ffffffu) | (2u << 30);
      asm volatile("tensor_load_to_lds %0, %1" :: "s"(g0), "s"(g1) : "memory");
      __builtin_amdgcn_s_wait_tensorcnt(1);  // oldest (row t) complete
    } else {
      __builtin_amdgcn_s_wait_tensorcnt(0);
    }

    float lpv[9];
#pragma unroll
    for (int j = 0; j < 8; ++j) lpv[j] = buf[cur][8 * l + j];
    lpv[8] = buf[cur][256];                  // only meaningful on lane 31

    float up1 = __shfl_up(a[7], 1, 32);      // state 8l-1 (prev lane top)
    float up2 = __shfl_up(a[6], 1, 32);      // state 8l-2
    if (l == 0) { up1 = NEGF; up2 = NEGF; }

    float na[9];
#pragma unroll
    for (int j = 0; j < 9; ++j) {
      float a1v = a[j];
      float a2v = (j == 0) ? up1 : a[j - 1];
      float a3v = ((j == 0) ? up2 : (j == 1) ? up1 : a[j - 2]) + sk[j];
      float mx  = fmaxf(fmaxf(a1v, a2v), a3v);
      float sum = __expf(a1v - mx) + __expf(a2v - mx) + __expf(a3v - mx);
      na[j] = mx + __logf(sum) + lpv[j];
    }
    const bool active = (t < hl);            // freeze alpha past h_len
#pragma unroll
    for (int j = 0; j < 9; ++j) a[j] = active ? na[j] : a[j];
  }

  // readout: loss_b = -logaddexp(alpha[2*yl], alpha[2*yl-1]) / yl
#pragma unroll
  for (int j = 0; j < 8; ++j) afin[8 * l + j] = a[j];
  if (l == 31) afin[256] = a[8];
  __syncthreads();
  if (l == 0) {
    int yl = ylen[b];
    int ib = 2 * yl;
    float x  = afin[ib];
    float y  = afin[ib - 1];
    float mx = fmaxf(x, y);
    float lae = mx + __logf(__expf(x - mx) + __expf(y - mx));
    lossb[b] = -lae / (float)yl;
  }
}

// ---------------------------------------------------------------------------
// Final scalar: mean over batch.
// ---------------------------------------------------------------------------
__global__ __launch_bounds__(32) void ctc_reduce(
    const float* __restrict__ lossb, float* __restrict__ out)
{
  int l = threadIdx.x;
  float v = (l < BB) ? lossb[l] : 0.0f;
#pragma unroll
  for (int off = 16; off >= 1; off >>= 1) v += __shfl_xor(v, off, 32);
  if (l == 0) out[0] = v / (float)BB;
}

extern "C" void kernel_launch(void* const* d_in, const int* in_sizes, int n_in,
                              void* d_out, int out_size, void* d_ws, size_t ws_size,
                              hipStream_t stream)
{
  (void)in_sizes; (void)n_in; (void)out_size; (void)ws_size;
  const float* hs   = (const float*)d_in[0];
  const int*   hlen = (const int*)d_in[1];
  const int*   ys   = (const int*)d_in[2];
  const int*   ylen = (const int*)d_in[3];
  float* out = (float*)d_out;

  char* ws = (char*)d_ws;
  float* lp    = (float*)ws;                                            // B*T*SP
  float* skipm = (float*)(ws + (size_t)BB * TT * SP * 4);               // B*SP
  float* lossb = (float*)(ws + (size_t)BB * TT * SP * 4
                             + (size_t)BB * SP * 4);                    // B

  ctc_lse_gather<<<(BB * TT) / 8, 256, 0, stream>>>(hs, ys, lp, skipm);
  ctc_alpha_dp<<<BB, 32, 0, stream>>>(lp, skipm, hlen, ylen, lossb);
  ctc_reduce<<<1, 32, 0, stream>>>(lossb, out);
}